// GCN_32487132627312
// MI455X (gfx1250) — compile-verified
//
#include <hip/hip_runtime.h>

// ---------------------------------------------------------------------------
// 3-layer GCN for MI455X (gfx1250, wave32).
//   GEMMs         : V_WMMA_F32_16X16X4_F32 (exact fp32 math).
//   Aggregation   : CSR (built once per launch) + pull-style wave-per-node
//                   gather with fused self-loop/bias/ReLU -> no f32 atomic
//                   RMW storms, ~2x less L2 traffic than push+scatter.
//   h matrix = 25.6MB << 192MB L2, so all gather traffic stays on-chip.
// ---------------------------------------------------------------------------

#define N_NODES    50000
#define N_EDGES    800000
#define D_IN       128
#define D_HID      128
#define D_OUT      64
#define NUM_GRAPHS 64
#define SCAN_THREADS 1024

typedef __attribute__((ext_vector_type(2))) float v2f;
typedef __attribute__((ext_vector_type(8))) float v8f;

// ---- in-degree count (no self loop; self loop handled analytically) -------
__global__ void degcount_kernel(const int* __restrict__ dst, int* __restrict__ deg, int E) {
    int e = blockIdx.x * blockDim.x + threadIdx.x;
    if (e < E) atomicAdd(&deg[dst[e]], 1);
}

__global__ void dis_kernel(const int* __restrict__ deg, float* __restrict__ dis, int n) {
    int i = blockIdx.x * blockDim.x + threadIdx.x;
    if (i < n) dis[i] = rsqrtf((float)deg[i] + 1.0f);   // deg + self loop
}

// ---- single-block exclusive scan over deg -> rowPtr[0..n] -----------------
__global__ void scan_kernel(const int* __restrict__ deg, int* __restrict__ rowPtr, int n) {
    __shared__ int sm[SCAN_THREADS];
    __shared__ int s_running;
    const int tid = threadIdx.x;
    if (tid == 0) s_running = 0;
    __syncthreads();
    for (int base = 0; base < n; base += SCAN_THREADS) {
        int idx = base + tid;
        int v = (idx < n) ? deg[idx] : 0;
        sm[tid] = v;
        __syncthreads();
        int acc = v;
        for (int off = 1; off < SCAN_THREADS; off <<= 1) {
            int t = (tid >= off) ? sm[tid - off] : 0;
            __syncthreads();
            acc += t;
            sm[tid] = acc;
            __syncthreads();
        }
        int run = s_running;                    // stable: written only below
        if (idx < n) rowPtr[idx] = run + acc - v;   // exclusive
        __syncthreads();
        if (tid == SCAN_THREADS - 1) s_running = run + sm[SCAN_THREADS - 1];
        __syncthreads();
    }
    if (tid == 0) rowPtr[n] = s_running;        // == E
}

// ---- bucket edges by dst, precompute norm = dis[src]*dis[dst] -------------
__global__ void reorder_kernel(const int* __restrict__ src, const int* __restrict__ dst,
                               const float* __restrict__ dis,
                               const int* __restrict__ rowPtr, int* __restrict__ cursor,
                               int* __restrict__ eSrc, float* __restrict__ eNrm, int E) {
    int e = blockIdx.x * blockDim.x + threadIdx.x;
    if (e >= E) return;
    int s = src[e];
    int d = dst[e];
    int pos = rowPtr[d] + atomicAdd(&cursor[d], 1);
    eSrc[pos] = s;
    eNrm[pos] = dis[s] * dis[d];
}

// ---- fp32 WMMA GEMM: out[nRows x DOUT] = A[nRows x K] @ W[K x DOUT] -------
// One wave per 16x16 output tile. K stepped by 4 with v_wmma_f32_16x16x4_f32.
// A 16x4 layout : lane L holds A[L%16][2*(L/16)+v] in reg v  -> float2 load
// B 4x16 layout : lane L holds B[2*(L/16)+v][L%16] in reg v
// C/D 16x16     : reg r, lane L -> row r + 8*(L/16), col L%16
template<int K, int DOUT>
__global__ void gemm_wmma_kernel(const float* __restrict__ A,
                                 const float* __restrict__ W,
                                 float* __restrict__ out, int nRows) {
    const int lane   = threadIdx.x & 31;
    const int waveId = threadIdx.x >> 5;
    const int tile   = blockIdx.x * (blockDim.x >> 5) + waveId;
    const int colTiles = DOUT / 16;
    const int rowTiles = nRows / 16;          // N_NODES % 16 == 0
    if (tile >= rowTiles * colTiles) return;  // wave-uniform: EXEC stays all-1s

    const int rowTile = tile / colTiles;
    const int colTile = tile - rowTile * colTiles;
    const int m0 = rowTile * 16;
    const int n0 = colTile * 16;

    const int m    = lane & 15;
    const int hi   = lane >> 4;     // 0 for lanes 0-15, 1 for lanes 16-31
    const int koff = hi * 2;        // K sub-offset this half-wave owns

    const float* arow = A + (size_t)(m0 + m) * K + koff;          // contiguous pairs
    const float* wcol = W + (size_t)koff * DOUT + n0 + (lane & 15);

    v8f acc = {0.f, 0.f, 0.f, 0.f, 0.f, 0.f, 0.f, 0.f};
#pragma unroll 8
    for (int k = 0; k < K; k += 4) {
        v2f a = *(const v2f*)(arow + k);
        v2f b;
        b.x = wcol[(size_t)k * DOUT];
        b.y = wcol[(size_t)(k + 1) * DOUT];
        acc = __builtin_amdgcn_wmma_f32_16x16x4_f32(
            /*neg_a=*/false, a, /*neg_b=*/false, b,
            /*c_mod=*/(short)0, acc, /*reuse_a=*/false, /*reuse_b=*/false);
    }

    const int outRow0 = m0 + hi * 8;
    const int outCol  = n0 + (lane & 15);
#pragma unroll
    for (int r = 0; r < 8; ++r)
        out[(size_t)(outRow0 + r) * DOUT + outCol] = acc[r];
}

// ---- fused pull aggregation + self loop + bias (+ReLU) --------------------
// out[n] = sum_{e in CSR[n]} h[eSrc[e]] * eNrm[e]  +  h[n]*dis[n]^2 + b
// One wave per node; lane covers VEC consecutive features -> each neighbor
// row is one coalesced 512B (or 256B) wave load. Next row prefetched to hide
// the dependent eSrc->h load chain (global_prefetch_b8).
template<int D, int VEC>
__global__ void gcn_aggregate_kernel(const float* __restrict__ h,
                                     const int* __restrict__ rowPtr,
                                     const int* __restrict__ eSrc,
                                     const float* __restrict__ eNrm,
                                     const float* __restrict__ dis,
                                     const float* __restrict__ bias,
                                     float* __restrict__ out,
                                     int n, int doRelu) {
    const int lane = threadIdx.x & 31;
    const int node = (int)((blockIdx.x * blockDim.x + threadIdx.x) >> 5);
    if (node >= n) return;                    // wave-uniform exit

    const int f0 = lane * VEC;
    const float di = dis[node];
    const float selfw = di * di;
    const float* hn = h + (size_t)node * D + f0;

    float acc[VEC];
#pragma unroll
    for (int j = 0; j < VEC; ++j)
        acc[j] = hn[j] * selfw + bias[f0 + j];

    const int beg = rowPtr[node];
    const int end = rowPtr[node + 1];
    for (int i = beg; i < end; ++i) {
        const int s = eSrc[i];
        const float w = eNrm[i];
        const float* hs = h + (size_t)s * D + f0;
        if (i + 1 < end) {                    // prefetch next neighbor's row
            const int sn = eSrc[i + 1];
            __builtin_prefetch(h + (size_t)sn * D + f0, 0, 1);
        }
#pragma unroll
        for (int j = 0; j < VEC; ++j)
            acc[j] = fmaf(hs[j], w, acc[j]);
    }

    float* op = out + (size_t)node * D + f0;
#pragma unroll
    for (int j = 0; j < VEC; ++j)
        op[j] = doRelu ? fmaxf(acc[j], 0.0f) : acc[j];
}

// ---- mean pool over graphs ------------------------------------------------
__global__ void pool_accum_kernel(const float* __restrict__ feat,
                                  const int* __restrict__ batch,
                                  float* __restrict__ sums, int n) {
    int idx = blockIdx.x * blockDim.x + threadIdx.x;
    if (idx >= n * D_OUT) return;
    int node = idx / D_OUT;
    int f    = idx - node * D_OUT;
    atomicAdd(&sums[batch[node] * D_OUT + f], feat[idx]);
}

__global__ void pool_cnt_kernel(const int* __restrict__ batch,
                                float* __restrict__ cnt, int n) {
    int i = blockIdx.x * blockDim.x + threadIdx.x;
    if (i < n) atomicAdd(&cnt[batch[i]], 1.0f);
}

__global__ void pool_final_kernel(const float* __restrict__ sums,
                                  const float* __restrict__ cnt,
                                  float* __restrict__ out) {
    int idx = blockIdx.x * blockDim.x + threadIdx.x;
    if (idx >= NUM_GRAPHS * D_OUT) return;
    out[idx] = sums[idx] / fmaxf(cnt[idx / D_OUT], 1.0f);
}

// ---------------------------------------------------------------------------
extern "C" void kernel_launch(void* const* d_in, const int* in_sizes, int n_in,
                              void* d_out, int out_size, void* d_ws, size_t ws_size,
                              hipStream_t stream) {
    const float* x     = (const float*)d_in[0];
    const int*   ei    = (const int*)  d_in[1];     // [2, E] flat
    const int*   batch = (const int*)  d_in[2];
    const float* W1    = (const float*)d_in[3];
    const float* b1    = (const float*)d_in[4];
    const float* W2    = (const float*)d_in[5];
    const float* b2    = (const float*)d_in[6];
    const float* W3    = (const float*)d_in[7];
    const float* b3    = (const float*)d_in[8];

    const int* src = ei;
    const int* dst = ei + N_EDGES;

    // ---- workspace layout (all 16B-aligned by construction) ----
    char* p = (char*)d_ws;
    float* dis    = (float*)p;                 p += (size_t)N_NODES * 4;
    int*   degI   = (int*)p;                   p += (size_t)N_NODES * 4;
    int*   rowPtr = (int*)p;                   p += (size_t)(N_NODES + 4) * 4;
    int*   cursor = (int*)p;                   p += (size_t)N_NODES * 4;
    int*   eSrc   = (int*)p;                   p += (size_t)N_EDGES * 4;
    float* eNrm   = (float*)p;                 p += (size_t)N_EDGES * 4;
    float* bufA   = (float*)p;                 p += (size_t)N_NODES * D_HID * 4;  // h = X@W
    float* bufC   = (float*)p;                 p += (size_t)N_NODES * D_HID * 4;  // activation
    float* psum   = (float*)p;                 p += (size_t)NUM_GRAPHS * D_OUT * 4;
    float* pcnt   = (float*)p;

    float* feat = (float*)d_out;                       // [N, 64]
    float* pool = feat + (size_t)N_NODES * D_OUT;      // [64, 64]

    const int nodeWaveBlocks = (N_NODES + 7) / 8;      // wave-per-node, 8 waves/block

    // ---- CSR build (once per launch) ----
    hipMemsetAsync(degI,   0, (size_t)N_NODES * sizeof(int), stream);
    hipMemsetAsync(cursor, 0, (size_t)N_NODES * sizeof(int), stream);
    degcount_kernel<<<(N_EDGES + 255) / 256, 256, 0, stream>>>(dst, degI, N_EDGES);
    dis_kernel<<<(N_NODES + 255) / 256, 256, 0, stream>>>(degI, dis, N_NODES);
    scan_kernel<<<1, SCAN_THREADS, 0, stream>>>(degI, rowPtr, N_NODES);
    reorder_kernel<<<(N_EDGES + 255) / 256, 256, 0, stream>>>(
        src, dst, dis, rowPtr, cursor, eSrc, eNrm, N_EDGES);

    // ---- layer 1: relu(gcn(x, W1, b1)) -> bufC ----
    {
        int tiles = (N_NODES / 16) * (D_HID / 16);
        gemm_wmma_kernel<D_IN, D_HID><<<(tiles + 7) / 8, 256, 0, stream>>>(x, W1, bufA, N_NODES);
        gcn_aggregate_kernel<D_HID, 4><<<nodeWaveBlocks, 256, 0, stream>>>(
            bufA, rowPtr, eSrc, eNrm, dis, b1, bufC, N_NODES, 1);
    }

    // ---- layer 2: relu(gcn(bufC, W2, b2)) -> bufC ----
    {
        int tiles = (N_NODES / 16) * (D_HID / 16);
        gemm_wmma_kernel<D_HID, D_HID><<<(tiles + 7) / 8, 256, 0, stream>>>(bufC, W2, bufA, N_NODES);
        gcn_aggregate_kernel<D_HID, 4><<<nodeWaveBlocks, 256, 0, stream>>>(
            bufA, rowPtr, eSrc, eNrm, dis, b2, bufC, N_NODES, 1);
    }

    // ---- layer 3: gcn(bufC, W3, b3) -> features (d_out) ----
    {
        int tiles = (N_NODES / 16) * (D_OUT / 16);
        gemm_wmma_kernel<D_HID, D_OUT><<<(tiles + 7) / 8, 256, 0, stream>>>(bufC, W3, bufA, N_NODES);
        gcn_aggregate_kernel<D_OUT, 2><<<nodeWaveBlocks, 256, 0, stream>>>(
            bufA, rowPtr, eSrc, eNrm, dis, b3, feat, N_NODES, 0);
    }

    // ---- mean pool per graph ----
    hipMemsetAsync(psum, 0, (size_t)(NUM_GRAPHS * D_OUT + NUM_GRAPHS) * sizeof(float), stream);
    pool_accum_kernel<<<(N_NODES * D_OUT + 255) / 256, 256, 0, stream>>>(feat, batch, psum, N_NODES);
    pool_cnt_kernel<<<(N_NODES + 255) / 256, 256, 0, stream>>>(batch, pcnt, N_NODES);
    pool_final_kernel<<<(NUM_GRAPHS * D_OUT + 255) / 256, 256, 0, stream>>>(psum, pcnt, pool);
}